// S4Conv1D_46291157516316
// MI455X (gfx1250) — compile-verified
//
#include <hip/hip_runtime.h>

// S4 conv kernel for MI455X (gfx1250, wave32).
// Pipeline: [1-wave setup: Gauss-Jordan 32x32, Ad^L by 15 squarings, power
// iteration for top singular pair, Sherman-Morrison scalars] -> [per-frequency
// Kc via closed-form rank-1 update, written bit-reversed] -> [15x radix-2
// inverse-FFT stages in global ws] -> [Kd] -> [WMMA f32 16x16x4 GEMV over x]
// -> [deterministic 64-way reduce]. Bandwidth floor: 64MB of x / 23.3TB/s.

#define F_DIM 512
#define L_LEN 32768
#define N_DIM 32
#define DELTA 1e-3f
#define SLICES 64           // L-split for the GEMV (2048 waves total)

typedef __attribute__((ext_vector_type(2))) float v2f;
typedef __attribute__((ext_vector_type(8))) float v8f;

__device__ __forceinline__ float2 cmulf(float2 a, float2 b) {
  return make_float2(a.x * b.x - a.y * b.y, a.x * b.y + a.y * b.x);
}
__device__ __forceinline__ float2 cdivf(float2 a, float2 b) {
  float d = b.x * b.x + b.y * b.y;
  return make_float2((a.x * b.x + a.y * b.y) / d, (a.y * b.x - a.x * b.y) / d);
}

// ---------------------------------------------------------------------------
// Kernel 1: one wave (32 threads). All N=32 linear algebra + the 9 scalars
// needed by the Sherman-Morrison closed form of inv(s*11^T - diag(A)).
// ---------------------------------------------------------------------------
__global__ void s4_precompute(const float* __restrict__ A,
                              const float* __restrict__ Bv,
                              const float* __restrict__ Cv,
                              float* __restrict__ scal) {
  __shared__ float aug[N_DIM][2 * N_DIM + 1];
  __shared__ float E[N_DIM][N_DIM + 1];
  __shared__ float vv[N_DIM], wv[N_DIM], tv[N_DIM];
  __shared__ float Pv[N_DIM], Qv[N_DIM], Cts[N_DIM];
  const int r = threadIdx.x;  // one row per lane

  // Augmented [I - (d/2)A | I + (d/2)A]; solving gives Ad = inv(.)*(.)
  for (int c = 0; c < N_DIM; ++c) {
    float a = A[r * N_DIM + c];
    float eye = (r == c) ? 1.0f : 0.0f;
    aug[r][c] = eye - 0.5f * DELTA * a;
    aug[r][N_DIM + c] = eye + 0.5f * DELTA * a;
  }
  __syncthreads();
  // Gauss-Jordan; I - (d/2)A is strongly diagonally dominant -> no pivoting.
  for (int p = 0; p < N_DIM; ++p) {
    if (r == p) {
      float inv = 1.0f / aug[p][p];
      for (int c = 0; c < 2 * N_DIM; ++c) aug[p][c] *= inv;
    }
    __syncthreads();
    if (r != p) {
      float f = aug[r][p];
      for (int c = p; c < 2 * N_DIM; ++c) aug[r][c] -= f * aug[p][c];
    }
    __syncthreads();
  }
  // Elementwise Ad ** L via 15 squarings (L = 2^15).
  for (int c = 0; c < N_DIM; ++c) {
    float e = aug[r][N_DIM + c];
#pragma unroll
    for (int k = 0; k < 15; ++k) e *= e;
    E[r][c] = e;
  }
  __syncthreads();
  // Cts[n] = C[n] - sum_m C[m] * E[m][n]
  {
    float s = 0.f;
    for (int m = 0; m < N_DIM; ++m) s += Cv[m] * E[m][r];
    Cts[r] = Cv[r] - s;
  }
  // Power iteration on A^T A for the top singular pair (replaces SVD; the
  // joint sign ambiguity of (u,v) cancels in the gauge-invariant combos).
  vv[r] = 1.0f + 0.001f * (float)r;
  __syncthreads();
  for (int it = 0; it < 300; ++it) {
    float s = 0.f;
    for (int c = 0; c < N_DIM; ++c) s += A[r * N_DIM + c] * vv[c];
    wv[r] = s;
    __syncthreads();
    s = 0.f;
    for (int c = 0; c < N_DIM; ++c) s += A[c * N_DIM + r] * wv[c];
    tv[r] = s;
    __syncthreads();
    float nrm = 0.f;
    for (int c = 0; c < N_DIM; ++c) nrm += tv[c] * tv[c];
    vv[r] = tv[r] * rsqrtf(nrm);
    __syncthreads();
  }
  {
    float s = 0.f;
    for (int c = 0; c < N_DIM; ++c) s += A[r * N_DIM + c] * vv[c];
    wv[r] = s;
    __syncthreads();
    float sv2 = 0.f;
    for (int c = 0; c < N_DIM; ++c) sv2 += wv[c] * wv[c];
    float sv = sqrtf(sv2);   // sigma_1
    float sq = sqrtf(sv);
    Pv[r] = (wv[r] / sv) * sq;  // U[:,0]*sqrt(s1)
    Qv[r] = vv[r] * sq;         // V[:,0]*sqrt(s1)
  }
  __syncthreads();
  if (r == 0) {
    // M_l = diag(-A_ii) + s_l*11^T ; Dinv_i = -1/A_ii. Sherman-Morrison:
    // x^T R y = x^T Dinv y - h * (x^T Dinv 1)(1^T Dinv y), h = s/(1+s*sigma).
    float sigma = 0, a1 = 0, a2 = 0, a3 = 0, q1 = 0, q2 = 0, q3 = 0, c1 = 0, b1 = 0;
    for (int i = 0; i < N_DIM; ++i) {
      float di = -1.0f / A[i * N_DIM + i];
      sigma += di;
      a1 += Cts[i] * di * Bv[i];
      a2 += Cts[i] * di;
      a3 += di * Bv[i];
      q1 += Qv[i] * di * Pv[i];
      q2 += Qv[i] * di;
      q3 += di * Pv[i];
      c1 += Cts[i] * di * Pv[i];
      b1 += Qv[i] * di * Bv[i];
    }
    scal[0] = sigma; scal[1] = a1; scal[2] = a2; scal[3] = a3;
    scal[4] = q1; scal[5] = q2; scal[6] = q3; scal[7] = c1; scal[8] = b1;
  }
}

// ---------------------------------------------------------------------------
// Kernel 2: per-frequency Kc via closed form; stored bit-reversed so the
// subsequent in-place DIT IFFT needs no separate permutation pass.
// ---------------------------------------------------------------------------
__global__ void s4_build_kc(const float* __restrict__ scal,
                            float2* __restrict__ kc) {
  int l = blockIdx.x * blockDim.x + threadIdx.x;
  float sigma = scal[0], a1 = scal[1], a2 = scal[2], a3 = scal[3];
  float q1 = scal[4], q2 = scal[5], q3 = scal[6], c1 = scal[7], b1 = scal[8];
  const float TWO_PI = 6.28318530717958647692f;
  float theta = (TWO_PI / (float)L_LEN) * (float)l;
  float sn, cs;
  sincosf(theta, &sn, &cs);
  float2 onepz = make_float2(1.f + cs, sn);
  float2 onemz = make_float2(1.f - cs, -sn);
  float2 s = cdivf(onemz, onepz);               // (1-z)/(1+z)
  s.x *= 2.f / DELTA;
  s.y *= 2.f / DELTA;
  float2 den = make_float2(1.f + sigma * s.x, sigma * s.y);
  float2 h = cdivf(s, den);                     // s/(1+s*sigma)
  float2 t1  = make_float2(a1 - a2 * a3 * h.x, -a2 * a3 * h.y);  // Cts R B
  float2 qrp = make_float2(q1 - q2 * q3 * h.x, -q2 * q3 * h.y);  // Q^T R P
  float2 cp  = make_float2(c1 - a2 * q3 * h.x, -a2 * q3 * h.y);  // Cts R P
  float2 qb  = make_float2(b1 - q2 * a3 * h.x, -q2 * a3 * h.y);  // Q^T R B
  float2 onepqrp = make_float2(1.f + qrp.x, qrp.y);
  float2 t2 = cdivf(cmulf(cp, qb), onepqrp);
  float2 diff = make_float2(t1.x - t2.x, t1.y - t2.y);
  float2 Kc = cdivf(make_float2(2.f * diff.x, 2.f * diff.y), onepz);
  unsigned rev = __brev((unsigned)l) >> 17;     // 15-bit bit-reversal
  kc[rev] = Kc;
}

// ---------------------------------------------------------------------------
// Kernel 3: one radix-2 DIT stage of the inverse FFT (sign +), in place.
// 16384 butterflies per stage; 15 launches total (~8MB traffic overall).
// ---------------------------------------------------------------------------
__global__ void s4_ifft_stage(float2* __restrict__ buf, int stage) {
  int t = blockIdx.x * blockDim.x + threadIdx.x;  // [0, L/2)
  int half = 1 << (stage - 1);
  int j = t & (half - 1);
  int grp = t >> (stage - 1);
  int base = grp * (half << 1) + j;
  float ang = (6.28318530717958647692f / (float)(half << 1)) * (float)j;
  float sn, cs;
  sincosf(ang, &sn, &cs);                         // w = e^{+i*2pi*j/m}
  float2 w = make_float2(cs, sn);
  float2 u = buf[base];
  float2 v = cmulf(buf[base + half], w);
  buf[base] = make_float2(u.x + v.x, u.y + v.y);
  buf[base + half] = make_float2(u.x - v.x, u.y - v.y);
}

__global__ void s4_finalize_kd(const float2* __restrict__ buf,
                               float* __restrict__ kd) {
  int l = blockIdx.x * blockDim.x + threadIdx.x;
  kd[l] = buf[l].x * (1.0f / (float)L_LEN);       // real(ifft)
}

// ---------------------------------------------------------------------------
// Kernel 4: out = Kd @ x via V_WMMA_F32_16X16X4_F32 (fp32-exact matrix pipe).
// Each wave: one 16-column tile of F x one 1/64 slice of L; A-fragment row 0
// carries Kd, rows 1..15 are zeroed via cndmask (loads are UNCONDITIONAL so
// no exec-masked branches appear in the hot loop). 32-bit A layout:
// VGPR0={K0|K2}, VGPR1={K1|K3} (lanes 0-15 | 16-31); B mirrored. D row 0
// lands in acc[0] on lanes 0-15. Partials reduced deterministically.
// ---------------------------------------------------------------------------
__global__ void s4_matvec_wmma(const float* __restrict__ kd,
                               const float* __restrict__ x,
                               float* __restrict__ partial) {
  int lane = threadIdx.x & 31;
  int wave = threadIdx.x >> 5;
  int gw = blockIdx.x * (blockDim.x >> 5) + wave;  // [0, 32*SLICES)
  int colTile = gw & 31;                           // F/16 = 32 tiles
  int slice = gw >> 5;                             // SLICES L-slices
  int f0 = colTile * 16;
  int lbase = slice * (L_LEN / SLICES);
  int laneM = lane & 15;
  int hi = lane >> 4;
  bool isRow0 = (laneM == 0);
  v8f acc = {};
  // Per-lane streaming pointers; advance by 4 rows per K-step.
  const float* kp = kd + lbase + 2 * hi;                           // b64/step
  const float* xp = x + (size_t)(lbase + 2 * hi) * F_DIM + f0 + laneM;
#pragma unroll 4
  for (int it = 0; it < (L_LEN / SLICES) / 4; ++it) {
    float2 kv = *(const float2*)kp;     // unconditional aligned load
    v2f a, b;
    a.x = isRow0 ? kv.x : 0.0f;         // v_cndmask, mask hoisted
    a.y = isRow0 ? kv.y : 0.0f;
    b.x = xp[0];                        // row r0   , 16 consecutive f
    b.y = xp[F_DIM];                    // row r0+1
    // Keep the x stream ahead of the loads (global_prefetch_b8; speculative,
    // silently dropped past the tail).
    __builtin_prefetch(xp + 8 * F_DIM, 0, 0);
    acc = __builtin_amdgcn_wmma_f32_16x16x4_f32(
        /*neg_a=*/false, a, /*neg_b=*/false, b,
        /*c_mod=*/(short)0, acc, /*reuse_a=*/false, /*reuse_b=*/false);
    kp += 4;
    xp += 4 * F_DIM;
  }
  if (lane < 16) partial[slice * F_DIM + f0 + lane] = acc[0];
}

__global__ void s4_reduce(const float* __restrict__ partial,
                          float* __restrict__ out) {
  int f = blockIdx.x * blockDim.x + threadIdx.x;
  if (f < F_DIM) {
    float s = 0.f;
#pragma unroll
    for (int k = 0; k < SLICES; ++k) s += partial[k * F_DIM + f];
    out[f] = s;
  }
}

extern "C" void kernel_launch(void* const* d_in, const int* in_sizes, int n_in,
                              void* d_out, int out_size, void* d_ws, size_t ws_size,
                              hipStream_t stream) {
  (void)in_sizes; (void)n_in; (void)out_size; (void)ws_size;
  const float* x  = (const float*)d_in[0];  // [L, F]
  const float* A  = (const float*)d_in[1];  // [N, N]
  const float* Bv = (const float*)d_in[2];  // [N, 1]
  const float* Cv = (const float*)d_in[3];  // [N, 1]
  float* out = (float*)d_out;               // [F]

  char* ws = (char*)d_ws;
  float2* kc   = (float2*)ws;                                 // 256 KB
  float*  kd   = (float*)(ws + 262144);                       // 128 KB
  float*  part = (float*)(ws + 262144 + 131072);              // SLICES*2KB
  float*  scal = (float*)(ws + 262144 + 131072 + SLICES * F_DIM * 4);

  s4_precompute<<<1, 32, 0, stream>>>(A, Bv, Cv, scal);
  s4_build_kc<<<L_LEN / 256, 256, 0, stream>>>(scal, kc);
  for (int s = 1; s <= 15; ++s)
    s4_ifft_stage<<<(L_LEN / 2) / 256, 256, 0, stream>>>(kc, s);
  s4_finalize_kd<<<L_LEN / 256, 256, 0, stream>>>(kc, kd);
  // 32 column tiles x SLICES L-slices = 2048 waves, 4 waves/block.
  s4_matvec_wmma<<<(32 * SLICES) / 4, 128, 0, stream>>>(kd, x, part);
  s4_reduce<<<2, 256, 0, stream>>>(part, out);
}